// GNN_13589276524780
// MI455X (gfx1250) — compile-verified
//
#include <hip/hip_runtime.h>
#include <hip/hip_bf16.h>

// ---------------------------------------------------------------------------
// GCN (2-layer) for MI455X / gfx1250.
//   deg/dis prep
//   -> [WMMA f32 GEMM h=X*W0, fused epilogue: agg=dis^2*h]
//   -> edge scatter (atomic f32, L2-resident)
//   -> [WMMA GEMM h=relu(agg)*W1, fused epilogue: out=dis^2*h]
//   -> edge scatter -> relu
// Whole working set (~115MB) is L2-resident (192MB); scatter is the limiter.
// ---------------------------------------------------------------------------

typedef __attribute__((ext_vector_type(2)))  float  v2f;
typedef __attribute__((ext_vector_type(8)))  float  v8f;

__device__ __forceinline__ void atomic_add_f32(float* p, float v) {
    // native global_atomic_add_f32 (RTNE per CDNA5 ISA ch.12)
    unsafeAtomicAdd(p, v);
}

// -------------------------- small elementwise kernels -----------------------

__global__ void __launch_bounds__(256) k_zero(float* __restrict__ p, int n) {
    int i = blockIdx.x * 256 + threadIdx.x;
    if (i < n) p[i] = 0.0f;
}

__global__ void __launch_bounds__(256) k_deg(const long long* __restrict__ ei,
                                             const float* __restrict__ ew,
                                             float* __restrict__ deg, int E) {
    int e = blockIdx.x * 256 + threadIdx.x;
    if (e >= E) return;
    long long d = ei[(long long)E + e];     // dst row of edge_index
    atomic_add_f32(&deg[d], ew[e]);
}

__global__ void __launch_bounds__(256) k_dis(const float* __restrict__ deg,
                                             float* __restrict__ dis, int n) {
    int i = blockIdx.x * 256 + threadIdx.x;
    if (i < n) dis[i] = rsqrtf(deg[i] + 1.0f);   // self-loop weight 1
}

__global__ void __launch_bounds__(256) k_relu(float* __restrict__ p, int n4) {
    int i = blockIdx.x * 256 + threadIdx.x;
    if (i >= n4) return;
    float4 v = ((float4*)p)[i];
    v.x = fmaxf(v.x, 0.f); v.y = fmaxf(v.y, 0.f);
    v.z = fmaxf(v.z, 0.f); v.w = fmaxf(v.w, 0.f);
    ((float4*)p)[i] = v;
}

// ------------------------- edge gather-scale-scatter ------------------------
// One wave per edge; 32 lanes x float2 cover the 64 feature columns.
// agg[dst] rows are 256B contiguous -> coalesced atomic bursts at L2.
__global__ void __launch_bounds__(256) k_scatter(const long long* __restrict__ ei,
                                                 const float* __restrict__ ew,
                                                 const float* __restrict__ dis,
                                                 const float* __restrict__ h,
                                                 float* __restrict__ agg, int E) {
    const int  lane = threadIdx.x & 31;
    const long long e = (long long)blockIdx.x * 8 + (threadIdx.x >> 5);
    if (e >= E) return;
    __builtin_prefetch(ei + e + 4096, 0, 0);     // global_prefetch_b8 on edge stream
    const long long s = ei[e];
    const long long d = ei[(long long)E + e];
    const float norm = dis[s] * ew[e] * dis[d];
    const float2 hv = *(const float2*)(h + s * 64 + lane * 2);
    float* dst = agg + d * 64 + lane * 2;
    atomic_add_f32(dst + 0, norm * hv.x);
    atomic_add_f32(dst + 1, norm * hv.y);
}

// ------------------------------ WMMA GEMM -----------------------------------
// H[n,64]  = (RELU ? relu(X) : X)[n,64] @ W[64,64]          (kept for scatter)
// AG[n,64] = dis[n]^2 * H[n,64]                             (self-loop init,
//            fused epilogue -> scatter target is fully initialized)
// 256 thr = 8 waves; each wave computes a 16x64 tile via V_WMMA_F32_16X16X4_F32,
// 16 K-steps, 4 column tiles. W staged once in LDS (16KB).
// n_rows must be a multiple of 16 (100000 is).
template <int RELU>
__global__ void __launch_bounds__(256, 2) k_gemm_wmma(const float* __restrict__ X,
                                                      const float* __restrict__ W,
                                                      const float* __restrict__ dis,
                                                      float* __restrict__ H,
                                                      float* __restrict__ AG,
                                                      int n_rows) {
    __shared__ float Wlds[64 * 64];
    for (int i = threadIdx.x; i < 64 * 64; i += 256) Wlds[i] = W[i];
    __syncthreads();

    const int lane = threadIdx.x & 31;
    const int wave = threadIdx.x >> 5;
    const int row_base = (blockIdx.x * 8 + wave) * 16;
    if (row_base >= n_rows) return;              // wave-uniform: EXEC stays all-1s

    const int m  = lane & 15;                    // row (A) / col-in-tile (B,D)
    const int kh = lane >> 4;                    // lane-half selects K sub-pair
    const float* xrow = X + (size_t)(row_base + m) * 64;

    v8f c0 = {}, c1 = {}, c2 = {}, c3 = {};

    for (int kk = 0; kk < 16; ++kk) {
        const int kb = kk * 4 + kh * 2;          // K pair for this lane half
        v2f a;
        a.x = xrow[kb + 0];
        a.y = xrow[kb + 1];
        if (RELU) { a.x = fmaxf(a.x, 0.f); a.y = fmaxf(a.y, 0.f); }
        v2f b0, b1, b2, b3;
        b0.x = Wlds[kb * 64 +  0 + m]; b0.y = Wlds[(kb + 1) * 64 +  0 + m];
        b1.x = Wlds[kb * 64 + 16 + m]; b1.y = Wlds[(kb + 1) * 64 + 16 + m];
        b2.x = Wlds[kb * 64 + 32 + m]; b2.y = Wlds[(kb + 1) * 64 + 32 + m];
        b3.x = Wlds[kb * 64 + 48 + m]; b3.y = Wlds[(kb + 1) * 64 + 48 + m];
        c0 = __builtin_amdgcn_wmma_f32_16x16x4_f32(false, a, false, b0, (short)0, c0, false, false);
        c1 = __builtin_amdgcn_wmma_f32_16x16x4_f32(false, a, false, b1, (short)0, c1, false, false);
        c2 = __builtin_amdgcn_wmma_f32_16x16x4_f32(false, a, false, b2, (short)0, c2, false, false);
        c3 = __builtin_amdgcn_wmma_f32_16x16x4_f32(false, a, false, b3, (short)0, c3, false, false);
    }

    // D layout: VGPR v -> row (row_base + 8*laneHalf + v), col = tile*16 + (lane&15)
    const int r0 = row_base + kh * 8;
    float* hrow = H  + (size_t)r0 * 64;
    float* arow = AG + (size_t)r0 * 64;
    #pragma unroll
    for (int v = 0; v < 8; ++v) {
        float s = dis[r0 + v];                   // broadcast within lane-half
        s *= s;
        float* ph = hrow + (size_t)v * 64;
        float* pa = arow + (size_t)v * 64;
        ph[ 0 + m] = c0[v];  pa[ 0 + m] = s * c0[v];
        ph[16 + m] = c1[v];  pa[16 + m] = s * c1[v];
        ph[32 + m] = c2[v];  pa[32 + m] = s * c2[v];
        ph[48 + m] = c3[v];  pa[48 + m] = s * c3[v];
    }
}

// ------------------------------- launcher -----------------------------------

extern "C" void kernel_launch(void* const* d_in, const int* in_sizes, int n_in,
                              void* d_out, int out_size, void* d_ws, size_t ws_size,
                              hipStream_t stream) {
    const float*     x  = (const float*)d_in[0];      // [N,64] f32
    const long long* ei = (const long long*)d_in[1];  // [2,E] int64
    const float*     ew = (const float*)d_in[2];      // [E]   f32
    const float*     W0 = (const float*)d_in[3];      // [64,64]
    const float*     W1 = (const float*)d_in[4];      // [64,64]
    float*           out = (float*)d_out;             // [N,64]

    const int N = in_sizes[0] / 64;
    const int E = in_sizes[2];

    // workspace: deg[N] | dis[N] | h[N*64] | agg[N*64]  (~52 MB)
    float* deg = (float*)d_ws;
    float* dis = deg + N;
    float* h   = dis + N;
    float* agg = h + (size_t)N * 64;

    const int nb_n    = (N + 255) / 256;
    const int nb_e    = (E + 255) / 256;
    const int nb_v4   = (N * 16 + 255) / 256;         // N*16 float4 elements
    const int nb_sc   = (E + 7) / 8;                  // 8 edges (waves) / block
    const int nb_gemm = ((N / 16) + 7) / 8;           // 8 row-tiles / block

    // degree + normalization
    k_zero<<<nb_n, 256, 0, stream>>>(deg, N);
    k_deg <<<nb_e, 256, 0, stream>>>(ei, ew, deg, E);
    k_dis <<<nb_n, 256, 0, stream>>>(deg, dis, N);

    // layer 1: h0 = X W0 ; agg1 = dis^2*h0 (fused) ; agg1 += scatter(norm*h0[src])
    k_gemm_wmma<0><<<nb_gemm, 256, 0, stream>>>(x, W0, dis, h, agg, N);
    k_scatter    <<<nb_sc,   256, 0, stream>>>(ei, ew, dis, h, agg, E);

    // layer 2: h1 = relu(agg1) W1 (relu fused in A-load) ; out = dis^2*h1 (fused)
    //          out += scatter(norm*h1[src]) ; out = relu(out)
    k_gemm_wmma<1><<<nb_gemm, 256, 0, stream>>>(agg, W1, dis, h, out, N);
    k_scatter    <<<nb_sc,   256, 0, stream>>>(ei, ew, dis, h, out, E);
    k_relu       <<<nb_v4,   256, 0, stream>>>(out, N * 16);
}